// ManifoldLoss_15109694947924
// MI455X (gfx1250) — compile-verified
//
#include <hip/hip_runtime.h>
#include <hip/hip_bf16.h>
#include <math.h>
#include <stdint.h>

#define IGNORE_INDEX (-1)

// ---------------------------------------------------------------------------
// Online logsumexp update for one accumulator stream.
// Common case (x <= m): 1 compare + 1 exp + 1 add, no dependence on other
// streams. Rare case (new max): rescale.
// ---------------------------------------------------------------------------
__device__ __forceinline__ void lse_step(float x, float& m, float& s) {
    if (x > m) {
        s = s * __expf(m - x) + 1.0f;
        m = x;
    } else {
        s += __expf(x - m);
    }
}

// Combine two (m, s) logsumexp partials; safe when either/both are (-inf, 0).
__device__ __forceinline__ void lse_combine(float& m, float& s, float om, float os) {
    float nm = fmaxf(m, om);
    float em = (m == nm)  ? 1.0f : __expf(m - nm);
    float eo = (om == nm) ? 1.0f : __expf(om - nm);
    s = s * em + os * eo;
    m = nm;
}

// ---------------------------------------------------------------------------
// Kernel 1: one 256-thread block (8 wave32) per row of V logits.
// Single pass over HBM (b128 loads), 4 independent accumulator streams for
// ILP. Produces per-row partials:
//   per_row = (1 - t)^2 + softplus(max_bg - t)
//   ce      = logsumexp(row) - t
//   valid   = (target != IGNORE_INDEX)
// ---------------------------------------------------------------------------
__global__ __launch_bounds__(256) void row_loss_kernel(
    const float* __restrict__ logits,
    const int*   __restrict__ targets,
    float* __restrict__ per_row,
    float* __restrict__ ce_row,
    int*   __restrict__ valid_row,
    int V)
{
    const int row = blockIdx.x;
    const float* __restrict__ p = logits + (size_t)row * (size_t)V;
    const int  tg    = targets[row];
    const bool valid = (tg != IGNORE_INDEX);
    const int  ts    = valid ? tg : 0;

    const int tid = threadIdx.x;

    // 4 independent streams: (max, sumexp, background-max) each.
    float m0 = -INFINITY, m1 = -INFINITY, m2 = -INFINITY, m3 = -INFINITY;
    float s0 = 0.f, s1 = 0.f, s2 = 0.f, s3 = 0.f;
    float b0 = -INFINITY, b1 = -INFINITY, b2 = -INFINITY, b3 = -INFINITY;

    // --- alignment prologue: rows start at (4*row*V) % 16 -> 0..3 elements ---
    const uintptr_t addr = (uintptr_t)p;            // always 4-byte aligned
    int pre = (int)(((16u - (unsigned)(addr & 15u)) & 15u) >> 2);
    if (pre > V) pre = V;
    for (int i = tid; i < pre; i += 256) {
        float x = p[i];
        lse_step(x, m0, s0);
        b0 = fmaxf(b0, (i == ts) ? -INFINITY : x);
    }

    // --- vector body: 16B-aligned b128 loads ---
    const int nv = (V - pre) >> 2;                  // number of float4's
    const float4* __restrict__ pv = (const float4*)(p + pre);

    // Which vec-iteration contains the target column (or -1 if none).
    const int i_t = (ts >= pre && ts < pre + (nv << 2)) ? ((ts - pre) >> 2) : -1;
    const int j_t = (ts - pre) & 3;                 // component within that vec4

    for (int i = tid; i < nv; i += 256) {
        // gfx1250 global_prefetch streaming hint, ~4 block-strides ahead
        int pf = i + 4 * 256;
        if (pf < nv) __builtin_prefetch(&pv[pf], 0, 1);

        float4 x = pv[i];
        lse_step(x.x, m0, s0);
        lse_step(x.y, m1, s1);
        lse_step(x.z, m2, s2);
        lse_step(x.w, m3, s3);

        if (i != i_t) {
            // hot path: plain max, no per-element index compare
            b0 = fmaxf(b0, x.x);
            b1 = fmaxf(b1, x.y);
            b2 = fmaxf(b2, x.z);
            b3 = fmaxf(b3, x.w);
        } else {
            // exactly one iteration per row: exclude the target component
            b0 = fmaxf(b0, (j_t == 0) ? -INFINITY : x.x);
            b1 = fmaxf(b1, (j_t == 1) ? -INFINITY : x.y);
            b2 = fmaxf(b2, (j_t == 2) ? -INFINITY : x.z);
            b3 = fmaxf(b3, (j_t == 3) ? -INFINITY : x.w);
        }
    }

    // --- scalar tail (0..3 elements) ---
    for (int i = pre + (nv << 2) + tid; i < V; i += 256) {
        float x = p[i];
        lse_step(x, m0, s0);
        b0 = fmaxf(b0, (i == ts) ? -INFINITY : x);
    }

    // --- merge the 4 streams ---
    lse_combine(m0, s0, m1, s1);
    lse_combine(m2, s2, m3, s3);
    lse_combine(m0, s0, m2, s2);
    float m   = m0, s = s0;
    float mbg = fmaxf(fmaxf(b0, b1), fmaxf(b2, b3));

    // --- wave32 reduction (gfx1250 is wave32-only) ---
    #pragma unroll
    for (int off = 16; off > 0; off >>= 1) {
        float om = __shfl_xor(m,   off, 32);
        float os = __shfl_xor(s,   off, 32);
        float ob = __shfl_xor(mbg, off, 32);
        lse_combine(m, s, om, os);
        mbg = fmaxf(mbg, ob);
    }

    // --- cross-wave reduction via LDS (8 waves per block) ---
    __shared__ float sm[8], ss[8], sb[8];
    const int lane = tid & 31;
    const int wave = tid >> 5;
    if (lane == 0) { sm[wave] = m; ss[wave] = s; sb[wave] = mbg; }
    __syncthreads();

    if (tid == 0) {
        float M = sm[0], S = ss[0], B = sb[0];
        #pragma unroll
        for (int w = 1; w < 8; ++w) {
            lse_combine(M, S, sm[w], ss[w]);
            B = fmaxf(B, sb[w]);
        }
        const float t   = p[ts];                  // target logit
        const float lse = M + logf(S);
        const float ce  = lse - t;

        // softplus(B - t), numerically stable (B = -inf -> 0)
        const float z  = B - t;
        const float sp = (z > 0.0f) ? (z + log1pf(expf(-z))) : log1pf(expf(z));

        const float d  = 1.0f - t;
        const float pr = d * d + sp;

        per_row[row]   = valid ? pr : 0.0f;
        ce_row[row]    = valid ? ce : 0.0f;
        valid_row[row] = valid ? 1 : 0;
    }
}

// ---------------------------------------------------------------------------
// Kernel 2: deterministic fixed-tree reduction of the N per-row partials.
// out = sum(per_row)/n_valid + (sum(ce)/n_valid) * 0.5
// ---------------------------------------------------------------------------
__global__ __launch_bounds__(1024) void finalize_kernel(
    const float* __restrict__ per_row,
    const float* __restrict__ ce_row,
    const int*   __restrict__ valid_row,
    int N,
    float* __restrict__ out)
{
    __shared__ float r1[1024];
    __shared__ float r2[1024];
    __shared__ int   r3[1024];
    const int tid = threadIdx.x;

    float a = 0.0f, b = 0.0f;
    int c = 0;
    for (int i = tid; i < N; i += 1024) {
        a += per_row[i];
        b += ce_row[i];
        c += valid_row[i];
    }
    r1[tid] = a; r2[tid] = b; r3[tid] = c;
    __syncthreads();

    #pragma unroll
    for (int off = 512; off > 0; off >>= 1) {
        if (tid < off) {
            r1[tid] += r1[tid + off];
            r2[tid] += r2[tid + off];
            r3[tid] += r3[tid + off];
        }
        __syncthreads();
    }

    if (tid == 0) {
        int nv = r3[0] > 1 ? r3[0] : 1;
        float inv = 1.0f / (float)nv;
        out[0] = r1[0] * inv + (r2[0] * inv) * 0.5f;
    }
}

// ---------------------------------------------------------------------------
// Host-side launcher (graph-capture safe: only kernel launches on `stream`).
// ---------------------------------------------------------------------------
extern "C" void kernel_launch(void* const* d_in, const int* in_sizes, int n_in,
                              void* d_out, int out_size, void* d_ws, size_t ws_size,
                              hipStream_t stream) {
    const float* logits  = (const float*)d_in[0];   // [B, S, V] f32
    const int*   targets = (const int*)d_in[1];     // [B, S] int32 (JAX x64 off)

    const int NV = in_sizes[0];     // B*S*V
    const int N  = in_sizes[1];     // B*S
    const int V  = NV / N;          // vocab

    // Workspace layout: per_row[N] f32 | ce_row[N] f32 | valid_row[N] i32
    float* per_row   = (float*)d_ws;
    float* ce_row    = per_row + N;
    int*   valid_row = (int*)(ce_row + N);

    row_loss_kernel<<<N, 256, 0, stream>>>(logits, targets,
                                           per_row, ce_row, valid_row, V);
    finalize_kernel<<<1, 1024, 0, stream>>>(per_row, ce_row, valid_row, N,
                                            (float*)d_out);
}